// OryxNetwork_14791867367617
// MI455X (gfx1250) — compile-verified
//
#include <hip/hip_runtime.h>
#include <math.h>

// Problem constants (match reference)
#define C_B    16
#define C_S    2048
#define C_E    256
#define C_H    8
#define C_A    16
#define C_D    32
#define C_OBS  128
#define C_CHUNK 64
#define C_NC   32
#define C_NBLK 2
#define C_P    (C_B * C_S)   // 32768 rows

typedef float v2f __attribute__((ext_vector_type(2)));
typedef float v8f __attribute__((ext_vector_type(8)));

__device__ __forceinline__ float gelu_f(float x) {
  float x3 = x * x * x;
  return 0.5f * x * (1.f + tanhf(0.7978845608028654f * (x + 0.044715f * x3)));
}
__device__ __forceinline__ float silu_f(float x) {
  return x / (1.f + expf(-x));
}
__device__ __forceinline__ v8f wmma4(v2f a, v2f b, v8f c) {
  return __builtin_amdgcn_wmma_f32_16x16x4_f32(false, a, false, b, (short)0, c, false, false);
}

// ---------------------------------------------------------------------------
// Tiled GEMM: C[M,N] = act(alpha * A[M,K] @ W[K,N] + bias[N])
// Block: 256 threads (8 waves), C tile 128x64, K chunks of 32 staged in LDS,
// double-buffered; chunk k+1 loaded to regs during compute, chunk k+2
// prefetched (global_prefetch_b8). Requires M%128==0, N%64==0, K%32==0.
// ---------------------------------------------------------------------------
#define GBM 128
#define GBN 64
#define GKC 32
#define GLDA 36
#define GLDB 68

template<int ACT>  // 0=none 1=gelu 2=silu
__global__ void gemm_tiled(const float* __restrict__ A, const float* __restrict__ W,
                           const float* __restrict__ bias, float* __restrict__ C,
                           int M, int K, int N, float alpha)
{
  __shared__ __align__(16) float sA[2][GBM * GLDA];
  __shared__ __align__(16) float sB[2][GKC * GLDB];
  const int tilesN = N / GBN;
  const int mBase = (blockIdx.x / tilesN) * GBM;
  const int nBase = (blockIdx.x % tilesN) * GBN;
  const int t = threadIdx.x;
  const int w = t >> 5;
  const int lane = t & 31;
  const int hh = lane >> 4, l = lane & 15;

  v8f acc[4];
  #pragma unroll
  for (int u = 0; u < 4; ++u)
    #pragma unroll
    for (int e = 0; e < 8; ++e) acc[u][e] = 0.f;

  const int nK = K / GKC;
  float4 ar[4], br[2];

  // prologue: chunk 0 -> regs -> LDS buf 0
  #pragma unroll
  for (int it = 0; it < 4; ++it) {
    const int p = t + 256 * it, r = p >> 3, c = (p & 7) * 4;
    ar[it] = *(const float4*)(A + (size_t)(mBase + r) * K + c);
  }
  #pragma unroll
  for (int it = 0; it < 2; ++it) {
    const int p = t + 256 * it, r = p >> 4, c = (p & 15) * 4;
    br[it] = *(const float4*)(W + (size_t)r * N + nBase + c);
  }
  #pragma unroll
  for (int it = 0; it < 4; ++it) {
    const int p = t + 256 * it, r = p >> 3, c = (p & 7) * 4;
    *(float4*)(&sA[0][r * GLDA + c]) = ar[it];
  }
  #pragma unroll
  for (int it = 0; it < 2; ++it) {
    const int p = t + 256 * it, r = p >> 4, c = (p & 15) * 4;
    *(float4*)(&sB[0][r * GLDB + c]) = br[it];
  }
  __syncthreads();

  for (int kc = 0; kc < nK; ++kc) {
    const int cur = kc & 1;
    const bool more = (kc + 1) < nK;
    if (more) {                       // issue next chunk's global loads early
      const int k0 = (kc + 1) * GKC;
      #pragma unroll
      for (int it = 0; it < 4; ++it) {
        const int p = t + 256 * it, r = p >> 3, c = (p & 7) * 4;
        ar[it] = *(const float4*)(A + (size_t)(mBase + r) * K + k0 + c);
      }
      #pragma unroll
      for (int it = 0; it < 2; ++it) {
        const int p = t + 256 * it, r = p >> 4, c = (p & 15) * 4;
        br[it] = *(const float4*)(W + (size_t)(k0 + r) * N + nBase + c);
      }
    }
    if (kc + 2 < nK) {                // hint chunk k+2 toward the caches
      const int k0p = (kc + 2) * GKC;
      #pragma unroll
      for (int it = 0; it < 4; ++it) {
        const int p = t + 256 * it, r = p >> 3, c = (p & 7) * 4;
        __builtin_prefetch(A + (size_t)(mBase + r) * K + k0p + c, 0, 3);
      }
      __builtin_prefetch(W + (size_t)(k0p + (t >> 4)) * N + nBase + (t & 15) * 4, 0, 3);
    }
    // compute current chunk from LDS
    const float* pA = &sA[cur][(16 * w + l) * GLDA];
    #pragma unroll
    for (int kk = 0; kk < GKC; kk += 4) {
      const v2f af = *(const v2f*)(pA + kk + 2 * hh);
      #pragma unroll
      for (int u = 0; u < 4; ++u) {
        const float* pb = &sB[cur][(kk + 2 * hh) * GLDB + u * 16 + l];
        v2f bf; bf.x = pb[0]; bf.y = pb[GLDB];
        acc[u] = wmma4(af, bf, acc[u]);
      }
    }
    __syncthreads();
    if (more) {
      const int nxt = cur ^ 1;
      #pragma unroll
      for (int it = 0; it < 4; ++it) {
        const int p = t + 256 * it, r = p >> 3, c = (p & 7) * 4;
        *(float4*)(&sA[nxt][r * GLDA + c]) = ar[it];
      }
      #pragma unroll
      for (int it = 0; it < 2; ++it) {
        const int p = t + 256 * it, r = p >> 4, c = (p & 15) * 4;
        *(float4*)(&sB[nxt][r * GLDB + c]) = br[it];
      }
      __syncthreads();
    }
  }

  #pragma unroll
  for (int u = 0; u < 4; ++u) {
    const int ncol = nBase + u * 16 + l;
    const float bv = bias ? bias[ncol] : 0.f;
    #pragma unroll
    for (int e = 0; e < 8; ++e) {
      float x = acc[u][e] * alpha + bv;
      if (ACT == 1)      x = gelu_f(x);
      else if (ACT == 2) x = silu_f(x);
      C[(size_t)(mBase + 16 * w + e + 8 * hh) * N + ncol] = x;
    }
  }
}

// ---------------------------------------------------------------------------
// Simple per-wave GEMM (for K=16 / N=16 cases). Wave computes 16x(16*NT).
// ---------------------------------------------------------------------------
template<int NT, int ACT>
__global__ void gemm_simple(const float* __restrict__ A, const float* __restrict__ W,
                            const float* __restrict__ bias, float* __restrict__ C,
                            int M, int K, int N, float alpha)
{
  const int wid  = blockIdx.x * (blockDim.x >> 5) + (threadIdx.x >> 5);
  const int lane = threadIdx.x & 31;
  const int tilesN = N / (16 * NT);
  const int totalTiles = (M >> 4) * tilesN;
  if (wid >= totalTiles) return;            // wave-uniform
  const int m0 = (wid / tilesN) * 16;
  const int n0 = (wid % tilesN) * (16 * NT);
  const int hh = lane >> 4, l = lane & 15;

  v8f acc[NT];
  #pragma unroll
  for (int u = 0; u < NT; ++u)
    #pragma unroll
    for (int e = 0; e < 8; ++e) acc[u][e] = 0.f;

  const float* __restrict__ arow = A + (size_t)(m0 + l) * K;
  for (int k0 = 0; k0 < K; k0 += 4) {
    const int ka = k0 + 2 * hh;
    v2f af; af.x = arow[ka]; af.y = arow[ka + 1];
    #pragma unroll
    for (int u = 0; u < NT; ++u) {
      const float* bp = W + (size_t)ka * N + (n0 + u * 16 + l);
      v2f bf; bf.x = bp[0]; bf.y = bp[N];
      acc[u] = wmma4(af, bf, acc[u]);
    }
  }
  #pragma unroll
  for (int u = 0; u < NT; ++u) {
    const int ncol = n0 + u * 16 + l;
    const float bv = bias ? bias[ncol] : 0.f;
    #pragma unroll
    for (int e = 0; e < 8; ++e) {
      float x = acc[u][e] * alpha + bv;
      if (ACT == 1)      x = gelu_f(x);
      else if (ACT == 2) x = silu_f(x);
      C[(size_t)(m0 + e + 8 * hh) * N + ncol] = x;
    }
  }
}

// ---------------------------------------------------------------------------
// RMSNorm over rows of length RL (optionally adds in1 first). One wave per row.
// ---------------------------------------------------------------------------
template<int RL>
__global__ void rmsnorm_kernel(const float* __restrict__ in0, const float* __restrict__ in1,
                               float* __restrict__ out)
{
  const int row  = blockIdx.x * 8 + (threadIdx.x >> 5);
  const int lane = threadIdx.x & 31;
  constexpr int PER = RL / 32;
  float vals[PER];
  float ss = 0.f;
  const size_t base = (size_t)row * RL;
  #pragma unroll
  for (int i = 0; i < PER; ++i) {
    float v = in0[base + i * 32 + lane];
    if (in1) v += in1[base + i * 32 + lane];
    vals[i] = v;
    ss += v * v;
  }
  #pragma unroll
  for (int o = 16; o > 0; o >>= 1) ss += __shfl_xor(ss, o);
  const float sc = rsqrtf(ss * (1.f / RL) + 1e-6f);
  #pragma unroll
  for (int i = 0; i < PER; ++i) out[base + i * 32 + lane] = vals[i] * sc;
}

// Fused: gate[i] *= groupnorm32(ret[i])  (mean/var over groups of 32).
__global__ void gnorm_gate_kernel(const float* __restrict__ ret, float* __restrict__ gate)
{
  const int row  = blockIdx.x * 8 + (threadIdx.x >> 5);
  const int lane = threadIdx.x & 31;
  const size_t idx = (size_t)row * 32 + lane;
  const float v = ret[idx];
  float s = v;
  #pragma unroll
  for (int o = 16; o > 0; o >>= 1) s += __shfl_xor(s, o);
  const float mu = s * (1.f / 32.f);
  const float d = v - mu;
  float s2 = d * d;
  #pragma unroll
  for (int o = 16; o > 0; o >>= 1) s2 += __shfl_xor(s2, o);
  gate[idx] = gate[idx] * (d * rsqrtf(s2 * (1.f / 32.f) + 1e-6f));
}

__global__ void mul_kernel(float* __restrict__ a, const float* __restrict__ b, int n)
{
  int i = blockIdx.x * blockDim.x + threadIdx.x;
  const int stride = gridDim.x * blockDim.x;
  for (; i < n; i += stride) a[i] *= b[i];
}

// ---------------------------------------------------------------------------
// Chunkwise retention scan, WMMA-based. One workgroup (8 waves) per (b,h).
// Masks are multiplies by 0/1 flags; the causal mask is folded into the
// symmetric decay table sKsym so every LDS load is unconditionally consumed
// (prevents the compiler from predicating loads with EXEC branches).
// q,k,v,out in (B,S,H*D) layout; h0 in (B,H,NBLK,D,D).
// ---------------------------------------------------------------------------
#define RS32 36   // padded stride for [*][32] LDS arrays (mult of 4: b128 stores)
#define RS64 66   // padded stride for [*][64] LDS arrays (even: b64 frag reads)

__global__ void retention_kernel(const float* __restrict__ q, const float* __restrict__ k,
                                 const float* __restrict__ v, const float* __restrict__ h0,
                                 const int* __restrict__ dones, float* __restrict__ out,
                                 int blk, int masked)
{
  const int bh = blockIdx.x;
  const int b = bh / C_H, h = bh % C_H;
  const int t = threadIdx.x;
  const int w = t >> 5, lane = t & 31;
  const int hh = lane >> 4, l = lane & 15;

  __shared__ __align__(16) float sQ[C_CHUNK * RS32];
  __shared__ __align__(16) float sK[C_CHUNK * RS32];
  __shared__ __align__(16) float sV[C_CHUNK * RS32];
  __shared__ __align__(16) float sKT[C_D * RS64];   // (k*kv_dec*keep)^T : [d][j]
  __shared__ __align__(16) float sAt[C_CHUNK * RS64];
  __shared__ __align__(16) float sH[C_D * RS32];
  __shared__ __align__(16) float sSeg[C_CHUNK];
  __shared__ __align__(16) float sKpow[C_CHUNK + 1];
  __shared__ __align__(16) float sKsym[128];        // sKsym[63+rel], causal mask folded in

  const float lg0 = logf(1.f / 32.f), lg1 = logf(1.f / 512.f);
  const float kap = (1.f - expf(lg0 + (float)h * (lg1 - lg0) * (1.f / 7.f))) * 0.9f;

  const float* h0p = h0 + (((size_t)(b * C_H + h)) * C_NBLK + blk) * (C_D * C_D);
  for (int f = t; f < C_D * C_D; f += 256) sH[(f >> 5) * RS32 + (f & 31)] = h0p[f];
  if (t <= C_CHUNK) sKpow[t] = powf(kap, (float)t);
  if (t < 128) {
    const int rel = t - 63;
    const int ar = rel < 0 ? -rel : rel;
    float val = powf(kap, (float)ar);
    if (masked && rel < 0) val = 0.f;
    sKsym[t] = val;
  }
  __syncthreads();
  const float chunk_dec = sKpow[C_CHUNK];

  for (int n = 0; n < C_NC; ++n) {
    const size_t rowBase = (size_t)b * C_S + (size_t)n * C_CHUNK;
    // ---- phase A: load chunk (b128) + segment cumsum ----
    #pragma unroll
    for (int it = 0; it < 2; ++it) {
      const int f = t + 256 * it;            // 512 float4s per matrix
      const int i = f >> 3, d4 = (f & 7) * 4;
      const size_t g = (rowBase + i) * C_E + h * C_D + d4;
      *(float4*)(&sQ[i * RS32 + d4]) = *(const float4*)(q + g);
      *(float4*)(&sK[i * RS32 + d4]) = *(const float4*)(k + g);
      *(float4*)(&sV[i * RS32 + d4]) = *(const float4*)(v + g);
    }
    if (t < C_CHUNK) {
      float s = 0.f;
      const int* dp = dones + (size_t)b * C_S + (size_t)n * C_CHUNK;
      for (int j = 0; j <= t; ++j) s += (dp[j] != 0) ? 1.f : 0.f;
      sSeg[t] = s;
    }
    __syncthreads();
    const float segLast = sSeg[C_CHUNK - 1];

    // ---- phase B: attn tiles (WMMA) + scaled transposed K ----
    #pragma unroll
    for (int tt = 0; tt < 2; ++tt) {
      const int tile = 2 * w + tt;
      const int m0 = (tile >> 2) * 16, n0 = (tile & 3) * 16;
      v8f acc;
      #pragma unroll
      for (int e = 0; e < 8; ++e) acc[e] = 0.f;
      const float* pa = &sQ[(m0 + l) * RS32];
      const float* pb = &sK[(n0 + l) * RS32];   // B = k^T: contiguous in k-row
      #pragma unroll
      for (int kk = 0; kk < C_D; kk += 4) {
        const v2f af = *(const v2f*)(pa + kk + 2 * hh);
        const v2f bf = *(const v2f*)(pb + kk + 2 * hh);
        acc = wmma4(af, bf, acc);
      }
      const int base = m0 + 8 * hh;            // multiple of 8 -> aligned float4
      const float4 sga = *(const float4*)(&sSeg[base]);
      const float4 sgb = *(const float4*)(&sSeg[base + 4]);
      const float segs[8] = {sga.x, sga.y, sga.z, sga.w, sgb.x, sgb.y, sgb.z, sgb.w};
      const int j = n0 + l;
      const float sj = sSeg[j];
      const float* psym = &sKsym[63 + base - j];
      #pragma unroll
      for (int e = 0; e < 8; ++e) {
        const float p = psym[e];                           // unconditional load
        const float same = (segs[e] == sj) ? 1.f : 0.f;    // cndmask of constants
        sAt[(base + e) * RS64 + j] = acc[e] * p * same;
      }
    }
    for (int f = t; f < C_CHUNK * C_D; f += 256) {
      const int j = f >> 5, d = f & 31;
      const float kp = sKpow[C_CHUNK - 1 - j];
      const float keep = (sSeg[j] == segLast) ? 1.f : 0.f;
      sKT[d * RS64 + j] = sK[j * RS32 + d] * kp * keep;
    }
    __syncthreads();

    // ---- phase C: out = attn@v + cross (1 tile/wave); kv on waves 0-3 ----
    {
      const int m0 = (w >> 1) * 16, n0 = (w & 1) * 16;
      v8f accI, accC;
      #pragma unroll
      for (int e = 0; e < 8; ++e) { accI[e] = 0.f; accC[e] = 0.f; }
      const float* pa = &sAt[(m0 + l) * RS64];
      #pragma unroll
      for (int kk = 0; kk < C_CHUNK; kk += 4) {
        const v2f af = *(const v2f*)(pa + kk + 2 * hh);
        const float* pv = &sV[(kk + 2 * hh) * RS32 + n0 + l];
        v2f bf; bf.x = pv[0]; bf.y = pv[RS32];
        accI = wmma4(af, bf, accI);
      }
      const float* pq = &sQ[(m0 + l) * RS32];
      #pragma unroll
      for (int kk = 0; kk < C_D; kk += 4) {
        const v2f af = *(const v2f*)(pq + kk + 2 * hh);
        const float* ph = &sH[(kk + 2 * hh) * RS32 + n0 + l];
        v2f bf; bf.x = ph[0]; bf.y = ph[RS32];
        accC = wmma4(af, bf, accC);
      }
      const int base = m0 + 8 * hh;
      const float4 sga = *(const float4*)(&sSeg[base]);
      const float4 sgb = *(const float4*)(&sSeg[base + 4]);
      const float segs[8] = {sga.x, sga.y, sga.z, sga.w, sgb.x, sgb.y, sgb.z, sgb.w};
      #pragma unroll
      for (int e = 0; e < 8; ++e) {
        const int i = base + e;
        const float fresh = (segs[e] == 0.f) ? 1.f : 0.f;
        const float cf = sKpow[i + 1] * fresh;
        out[(rowBase + i) * C_E + h * C_D + n0 + l] = accI[e] + cf * accC[e];
      }
    }
    v8f accKV;
    #pragma unroll
    for (int e = 0; e < 8; ++e) accKV[e] = 0.f;
    if (w < 4) {
      const int m0 = (w >> 1) * 16, n0 = (w & 1) * 16;
      const float* pa = &sKT[(m0 + l) * RS64];
      #pragma unroll
      for (int kk = 0; kk < C_CHUNK; kk += 4) {
        const v2f af = *(const v2f*)(pa + kk + 2 * hh);
        const float* pv = &sV[(kk + 2 * hh) * RS32 + n0 + l];
        v2f bf; bf.x = pv[0]; bf.y = pv[RS32];
        accKV = wmma4(af, bf, accKV);
      }
    }
    __syncthreads();   // cross reads of old sH complete

    // ---- phase D: h = h*chunk_dec*carry + kv ----
    if (w < 4) {
      const float carry = (segLast == 0.f) ? chunk_dec : 0.f;
      const int m0 = (w >> 1) * 16, n0 = (w & 1) * 16;
      #pragma unroll
      for (int e = 0; e < 8; ++e) {
        const int d = m0 + e + 8 * hh;
        sH[d * RS32 + n0 + l] = sH[d * RS32 + n0 + l] * carry + accKV[e];
      }
    }
    __syncthreads();
  }
}

// ---------------------------------------------------------------------------
// Host-side launch helpers
// ---------------------------------------------------------------------------
template<int ACT>
static inline void gemm_big(const float* A, const float* W, const float* bias, float* C,
                            int M, int K, int N, float alpha, hipStream_t s)
{
  const int blocks = (M / GBM) * (N / GBN);
  gemm_tiled<ACT><<<blocks, 256, 0, s>>>(A, W, bias, C, M, K, N, alpha);
}
template<int NT, int ACT>
static inline void gemm_small(const float* A, const float* W, const float* bias, float* C,
                              int M, int K, int N, float alpha, hipStream_t s)
{
  const int waves = (M >> 4) * (N / (16 * NT));
  gemm_simple<NT, ACT><<<(waves + 7) / 8, 256, 0, s>>>(A, W, bias, C, M, K, N, alpha);
}
static inline void rms128(const float* i0, const float* i1, float* o, int rows, hipStream_t s) {
  rmsnorm_kernel<128><<<rows / 8, 256, 0, s>>>(i0, i1, o);
}
static inline void rms256(const float* i0, const float* i1, float* o, int rows, hipStream_t s) {
  rmsnorm_kernel<256><<<rows / 8, 256, 0, s>>>(i0, i1, o);
}

struct Bufs { float *OBSR, *X, *Qb, *Kb, *Vb, *Rb, *Gb, *RNb; };

// msr(xk, xq, xv, hs[:, :, blk], dones, W[5], masked) -> outbuf (may alias Qb)
static void msr_call(const float* xk, const float* xq, const float* xv,
                     const float* hs, int blk, const int* dones,
                     const float* W5, int masked, Bufs& Bf, float* outbuf, hipStream_t s)
{
  const size_t WE = (size_t)C_E * C_E;
  gemm_big<0>(xq, W5 + 0 * WE, nullptr, Bf.Qb, C_P, C_E, C_E, 1.f, s);
  gemm_big<0>(xk, W5 + 1 * WE, nullptr, Bf.Kb, C_P, C_E, C_E, 0.17677669529663687f, s); // 1/sqrt(D)
  gemm_big<0>(xv, W5 + 2 * WE, nullptr, Bf.Vb, C_P, C_E, C_E, 1.f, s);
  retention_kernel<<<C_B * C_H, 256, 0, s>>>(Bf.Qb, Bf.Kb, Bf.Vb, hs, dones, Bf.Rb, blk, masked);
  gemm_big<2>(xq, W5 + 3 * WE, nullptr, Bf.Gb, C_P, C_E, C_E, 1.f, s);     // silu gate
  gnorm_gate_kernel<<<(C_P * C_H) / 8, 256, 0, s>>>(Bf.Rb, Bf.Gb);          // gate *= gn(ret)
  gemm_big<0>(Bf.Gb, W5 + 4 * WE, nullptr, outbuf, C_P, C_E, C_E, 1.f, s);
}

extern "C" void kernel_launch(void* const* d_in, const int* in_sizes, int n_in,
                              void* d_out, int out_size, void* d_ws, size_t ws_size,
                              hipStream_t stream)
{
  (void)in_sizes; (void)n_in; (void)out_size; (void)ws_size;
  const float* obs       = (const float*)d_in[0];
  const float* action    = (const float*)d_in[1];
  const float* hs_enc    = (const float*)d_in[2];
  const float* hs_dec1   = (const float*)d_in[3];
  const float* hs_dec2   = (const float*)d_in[4];
  const int*   dones     = (const int*)  d_in[5];
  const float* W_obs     = (const float*)d_in[6];
  const float* W_act     = (const float*)d_in[7];
  const float* enc_ret_W = (const float*)d_in[8];
  const float* enc_ffn_W = (const float*)d_in[9];
  const float* dec_ret1_W= (const float*)d_in[10];
  const float* dec_ret2_W= (const float*)d_in[11];
  const float* dec_ffn_W = (const float*)d_in[12];
  const float* head_W1   = (const float*)d_in[13];
  const float* head_b1   = (const float*)d_in[14];
  const float* head_W2   = (const float*)d_in[15];
  const float* head_b2   = (const float*)d_in[16];
  float* out = (float*)d_out;

  const size_t SZ = (size_t)C_P * C_E;       // one activation buffer: 32 MB
  float* ws = (float*)d_ws;
  Bufs Bf;
  Bf.OBSR = ws + 0 * SZ; Bf.X  = ws + 1 * SZ; Bf.Qb = ws + 2 * SZ; Bf.Kb = ws + 3 * SZ;
  Bf.Vb   = ws + 4 * SZ; Bf.Rb = ws + 5 * SZ; Bf.Gb = ws + 6 * SZ; Bf.RNb = ws + 7 * SZ;
  const size_t WE = (size_t)C_E * C_E;

  // obs_rep = gelu(rmsnorm(obs) @ W_obs)
  rms128(obs, nullptr, Bf.Qb, C_P, stream);
  gemm_big<1>(Bf.Qb, W_obs, nullptr, Bf.OBSR, C_P, C_OBS, C_E, 1.f, stream);

  // ---- encoder ----
  for (int i = 0; i < C_NBLK; ++i) {
    const float* W5 = enc_ret_W + (size_t)i * 5 * WE;
    const float* Wf = enc_ffn_W + (size_t)i * 3 * WE;
    rms256(Bf.OBSR, nullptr, Bf.X, C_P, stream);                       // x = rmsnorm(obs_rep)
    msr_call(Bf.X, Bf.X, Bf.X, hs_enc, i, dones, W5, /*masked=*/0, Bf, Bf.Qb, stream);
    rms256(Bf.X, Bf.Qb, Bf.X, C_P, stream);                            // x = rmsnorm(x + ret)
    gemm_big<2>(Bf.X, Wf + 0 * WE, nullptr, Bf.Gb, C_P, C_E, C_E, 1.f, stream);
    gemm_big<0>(Bf.X, Wf + 1 * WE, nullptr, Bf.Kb, C_P, C_E, C_E, 1.f, stream);
    mul_kernel<<<4096, 256, 0, stream>>>(Bf.Gb, Bf.Kb, C_P * C_E);
    gemm_big<0>(Bf.Gb, Wf + 2 * WE, nullptr, Bf.Qb, C_P, C_E, C_E, 1.f, stream);
    rms256(Bf.X, Bf.Qb, Bf.OBSR, C_P, stream);                         // obs_rep = rmsnorm(x + ffn)
  }

  // ---- decoder ----
  gemm_small<4, 1>(action, W_act, nullptr, Bf.Qb, C_P, C_A, C_E, 1.f, stream);
  rms256(Bf.Qb, nullptr, Bf.X, C_P, stream);                           // x = rmsnorm(gelu(action@W_act))

  for (int i = 0; i < C_NBLK; ++i) {
    const float* W5a = dec_ret1_W + (size_t)i * 5 * WE;
    const float* W5b = dec_ret2_W + (size_t)i * 5 * WE;
    const float* Wf  = dec_ffn_W  + (size_t)i * 3 * WE;
    msr_call(Bf.X, Bf.X, Bf.X, hs_dec1, i, dones, W5a, /*masked=*/1, Bf, Bf.Qb, stream);
    rms256(Bf.X, Bf.Qb, Bf.RNb, C_P, stream);                          // ret = rmsnorm(x + ret1)
    msr_call(Bf.RNb, Bf.OBSR, Bf.RNb, hs_dec2, i, dones, W5b, /*masked=*/1, Bf, Bf.Qb, stream);
    rms256(Bf.OBSR, Bf.Qb, Bf.Kb, C_P, stream);                        // y = rmsnorm(obs_rep + ret2)
    gemm_big<2>(Bf.Kb, Wf + 0 * WE, nullptr, Bf.Gb, C_P, C_E, C_E, 1.f, stream);
    gemm_big<0>(Bf.Kb, Wf + 1 * WE, nullptr, Bf.Vb, C_P, C_E, C_E, 1.f, stream);
    mul_kernel<<<4096, 256, 0, stream>>>(Bf.Gb, Bf.Vb, C_P * C_E);
    gemm_big<0>(Bf.Gb, Wf + 2 * WE, nullptr, Bf.Qb, C_P, C_E, C_E, 1.f, stream);
    rms256(Bf.Kb, Bf.Qb, Bf.X, C_P, stream);                           // x = rmsnorm(y + swiglu)
  }

  // ---- heads ----
  for (int j = 0; j < 2; ++j) {
    gemm_big<1>(Bf.X, head_W1 + (size_t)j * WE, head_b1 + (size_t)j * C_E,
                Bf.Qb, C_P, C_E, C_E, 1.f, stream);
    rms256(Bf.Qb, nullptr, Bf.Kb, C_P, stream);
    gemm_small<1, 0>(Bf.Kb, head_W2 + (size_t)j * C_E * C_A, head_b2 + (size_t)j * C_A,
                     out + (size_t)j * (size_t)C_P * C_A, C_P, C_E, C_A, 1.f, stream);
  }
}